// SSIM_4277787426942
// MI455X (gfx1250) — compile-verified
//
#include <hip/hip_runtime.h>

// ---------------- geometry ----------------
#define IMG_H 512
#define IMG_W 512
#define TW    128          // output tile width  (per block)
#define TH    16           // output tile height (per block) == WMMA M
#define PW    130          // padded tile width  (TW + 2)
#define PH    18           // padded tile height (TH + 2) == live H rows (K clamped)
#define PSTR  130          // LDS row stride of padded input tiles (floats)
#define HSTR  130          // LDS row stride of H buffers (floats)
#define HSZ   (PH * HSTR)  // one H buffer: 2340 floats

typedef float v2f __attribute__((ext_vector_type(2)));
typedef float v8f __attribute__((ext_vector_type(8)));

typedef __attribute__((address_space(1))) int as1_int;   // global AS ("__device__")
typedef __attribute__((address_space(3))) int as3_int;   // LDS AS

// ---------------- CDNA5 feature probes ----------------
#if defined(__has_builtin)
#  if __has_builtin(__builtin_amdgcn_global_load_async_to_lds_b32)
#    define HAVE_ASYNC_LDS 1
#  else
#    define HAVE_ASYNC_LDS 0
#  endif
#  if __has_builtin(__builtin_amdgcn_s_wait_asynccnt)
#    define HAVE_WAIT_ASYNC 1
#  else
#    define HAVE_WAIT_ASYNC 0
#  endif
#  if __has_builtin(__builtin_amdgcn_wmma_f32_16x16x4_f32)
#    define HAVE_WMMA_F32K4 1
#  else
#    define HAVE_WMMA_F32K4 0
#  endif
#else
#  define HAVE_ASYNC_LDS 0
#  define HAVE_WAIT_ASYNC 0
#  define HAVE_WMMA_F32K4 0
#endif

__global__ __launch_bounds__(256)
void ssim_fused_kernel(const float* __restrict__ x,
                       const float* __restrict__ y,
                       float* __restrict__ out)
{
    __shared__ float xs[PH * PSTR];     // reflect-padded x tile      (9360 B)
    __shared__ float ys[PH * PSTR];     // reflect-padded y tile      (9360 B)
    __shared__ float Hs[5 * HSZ];       // 5 horizontal 3-tap sums    (46800 B)
                                        // total 65520 B <= 64 KB

    const int tid  = threadIdx.x;
    const int lane = tid & 31;
    const int wave = tid >> 5;          // 0..7 -> 16-wide column group
    const int gx0  = blockIdx.x * TW;
    const int gy0  = blockIdx.y * TH;
    const size_t ibase = (size_t)blockIdx.z * (size_t)(IMG_H * IMG_W);
    const float* xb = x + ibase;
    const float* yb = y + ibase;

    // ---- Stage reflect-padded tiles: async global -> LDS (per-lane addrs) ----
    for (int idx = tid; idx < PH * PW; idx += 256) {
        int pr = idx / PW;
        int pc = idx - pr * PW;
        int gr = gy0 + pr - 1;
        gr = (gr < 0) ? -gr : gr;
        gr = (gr >= IMG_H) ? (2 * IMG_H - 2 - gr) : gr;
        int gc = gx0 + pc - 1;
        gc = (gc < 0) ? -gc : gc;
        gc = (gc >= IMG_W) ? (2 * IMG_W - 2 - gc) : gc;
        const float* gxp = xb + (size_t)gr * IMG_W + gc;
        const float* gyp = yb + (size_t)gr * IMG_W + gc;
        const int lofs = pr * PSTR + pc;
#if HAVE_ASYNC_LDS
        __builtin_amdgcn_global_load_async_to_lds_b32(
            (as1_int*)gxp, (as3_int*)&xs[lofs], 0, 0);
        __builtin_amdgcn_global_load_async_to_lds_b32(
            (as1_int*)gyp, (as3_int*)&ys[lofs], 0, 0);
#else
        xs[lofs] = __builtin_nontemporal_load(gxp);
        ys[lofs] = __builtin_nontemporal_load(gyp);
#endif
    }

#if HAVE_ASYNC_LDS
#  if HAVE_WAIT_ASYNC
    __builtin_amdgcn_s_wait_asynccnt(0);
#  else
    asm volatile("s_wait_asynccnt 0" ::: "memory");
#  endif
#endif
    __syncthreads();

    // ---- Phase A (single pass): all 5 horizontal 3-tap sums per position ----
    for (int p = tid; p < PH * TW; p += 256) {       // 2304 = exactly 9/thread
        const int r = p >> 7;                        // TW == 128
        const int c = p & (TW - 1);
        const int o = r * PSTR + c;
        const float x0 = xs[o], x1 = xs[o + 1], x2 = xs[o + 2];
        const float y0 = ys[o], y1 = ys[o + 1], y2 = ys[o + 2];
        const int hb = r * HSTR + c;
        Hs[0 * HSZ + hb] = x0 + x1 + x2;
        Hs[1 * HSZ + hb] = y0 + y1 + y2;
        Hs[2 * HSZ + hb] = x0 * x0 + x1 * x1 + x2 * x2;
        Hs[3 * HSZ + hb] = y0 * y0 + y1 * y1 + y2 * y2;
        Hs[4 * HSZ + hb] = x0 * y0 + x1 * y1 + x2 * y2;
    }
    __syncthreads();

    // ---- Constant banded A matrix (16x20, A[m][k] = k in [m, m+2]), K=4 chunks.
    // A-layout (32-bit A 16x4): lanes 0-15: M=lane, VGPR0=K0/VGPR1=K1; lanes 16-31: K2/K3.
    // A[m][18] = A[m][19] = 0 for all m, so B rows 18/19 are dead -> clamp to row 17.
    const int m     = lane & 15;
    const int khalf = (lane >> 4) << 1;     // 0 or 2
    v2f Aj[5];
#pragma unroll
    for (int j = 0; j < 5; ++j) {
        const int k0 = 4 * j + khalf;
        Aj[j].x = (k0     >= m && k0     <= m + 2) ? 1.0f : 0.0f;
        Aj[j].y = (k0 + 1 >= m && k0 + 1 <= m + 2) ? 1.0f : 0.0f;
    }
    const int cb = wave << 4;               // this wave's 16-column group

    // ---- Phase B: vertical 3-tap sums as banded matmuls on the matrix pipe ----
    v8f acc[5];
#pragma unroll
    for (int q = 0; q < 5; ++q) {
        const float* Hq = &Hs[q * HSZ];
        v8f d = {0.f, 0.f, 0.f, 0.f, 0.f, 0.f, 0.f, 0.f};
#if HAVE_WMMA_F32K4
#pragma unroll
        for (int j = 0; j < 5; ++j) {
            // B-layout (32-bit B 4x16): VGPR v, lanes 0-15: row v; lanes 16-31: row v+2.
            const int rb0 = 4 * j + khalf;
            const int rb1 = rb0 + 1;
            const int r0  = (rb0 > 17) ? 17 : rb0;   // dead rows (A==0) clamped in-bounds
            const int r1  = (rb1 > 17) ? 17 : rb1;
            v2f b;
            b.x = Hq[r0 * HSTR + cb + m];
            b.y = Hq[r1 * HSTR + cb + m];
            d = __builtin_amdgcn_wmma_f32_16x16x4_f32(
                    false, Aj[j], false, b, (short)0, d, false, false);
        }
#else
        {
            const int rsel = (lane >> 4) << 3;   // D VGPR e holds rows e and e+8
#pragma unroll
            for (int e = 0; e < 8; ++e) {
                const int r0 = e + rsel;
                d[e] = Hq[(r0)     * HSTR + cb + m]
                     + Hq[(r0 + 1) * HSTR + cb + m]
                     + Hq[(r0 + 2) * HSTR + cb + m];
            }
        }
#endif
        acc[q] = d;
    }

    // ---- SSIM formula, elementwise on the accumulator layout; store ----
    const float inv9 = 1.0f / 9.0f;
    const float C1v  = 1e-4f;     // 0.01^2
    const float C2v  = 9e-4f;     // 0.03^2
    const int rsel   = (lane >> 4) << 3;
    float* ob = out + ibase;
#pragma unroll
    for (int e = 0; e < 8; ++e) {
        const float mux = acc[0][e] * inv9;
        const float muy = acc[1][e] * inv9;
        const float sx  = acc[2][e] * inv9 - mux * mux;
        const float sy  = acc[3][e] * inv9 - muy * muy;
        const float sxy = acc[4][e] * inv9 - mux * muy;
        const float n   = (2.0f * mux * muy + C1v) * (2.0f * sxy + C2v);
        const float dd  = (mux * mux + muy * muy + C1v) * (sx + sy + C2v);
        float r = (1.0f - n / (dd + 1e-8f)) * 0.5f;
        r = fminf(fmaxf(r, 0.0f), 1.0f);
        const int orow = gy0 + e + rsel;
        const int ocol = gx0 + cb + m;
        ob[(size_t)orow * IMG_W + ocol] = r;
    }
}

extern "C" void kernel_launch(void* const* d_in, const int* in_sizes, int n_in,
                              void* d_out, int out_size, void* d_ws, size_t ws_size,
                              hipStream_t stream)
{
    (void)n_in; (void)out_size; (void)d_ws; (void)ws_size;
    const float* x = (const float*)d_in[0];
    const float* y = (const float*)d_in[1];
    float* out = (float*)d_out;
    const int nimg = in_sizes[0] / (IMG_H * IMG_W);   // B*C = 48
    dim3 grid(IMG_W / TW, IMG_H / TH, nimg);
    ssim_fused_kernel<<<grid, dim3(256), 0, stream>>>(x, y, out);
}